// ConfidenceWeightedLoss_52819507806484
// MI455X (gfx1250) — compile-verified
//
#include <hip/hip_runtime.h>

// CDNA5 / gfx1250. Wave32. Streaming bandwidth-bound reduction:
// 128 MB in -> 3 scalars (~5.5us HBM floor @ 23.3 TB/s).
// Hot loop: 8x nontemporal b128 loads + native v_exp_f32/v_log_f32/v_rcp_f32
// softmax-loss math (keeps VALU well under the memory bound).
// Wave reduction: one V_WMMA_F32_16X16X4_F32 folds both loss accumulators
// (32->16) in a single matrix op, then 4x ds_swizzle xor steps finish 16->1.

typedef float v2f __attribute__((ext_vector_type(2)));
typedef float v8f __attribute__((ext_vector_type(8)));
typedef float f4  __attribute__((ext_vector_type(4)));
typedef int   i4  __attribute__((ext_vector_type(4)));

#define CONF_THRESHOLD 0.8f
#define W_HC_CORRECT   0.3f
#define W_HC_WRONG     3.0f
#define W_LOW          1.0f
#define EPS_LOSS       1e-8f

// group-of-32 xor swizzle: offset[14:10]=xor_mask, offset[4:0]=and_mask=0x1F
#define XOR_SWIZZLE_ADD(v, mask)                                              \
    (v) += __int_as_float(__builtin_amdgcn_ds_swizzle(                        \
        __float_as_int(v), ((mask) << 10) | 0x1F))

__device__ __forceinline__ float row_loss(float l0, float l1, float l2, int label) {
    // argmax with first-max-wins (matches jnp.argmax tie rule)
    float m = l0; int pred = 0;
    if (l1 > m) { m = l1; pred = 1; }
    if (l2 > m) { m = l2; pred = 2; }
    // native transcendentals: v_exp_f32 / v_log_f32 / v_rcp_f32
    float e0 = __expf(l0 - m);
    float e1 = __expf(l1 - m);
    float e2 = __expf(l2 - m);
    float s   = e0 + e1 + e2;
    float inv = __builtin_amdgcn_rcpf(s);
    float e_pred = (pred == 0) ? e0 : (pred == 1) ? e1 : e2;   // == 1.0f
    float e_true = (label == 0) ? e0 : (label == 1) ? e1 : e2;
    float conf   = e_pred * inv;
    float p_true = e_true * inv;
    bool  correct = (pred == label);
    float wrong_w = (label == 1) ? (W_HC_WRONG * 2.0f) : W_HC_WRONG;
    float w = (conf > CONF_THRESHOLD) ? (correct ? W_HC_CORRECT : wrong_w) : W_LOW;
    return -w * __logf(p_true + EPS_LOSS);
}

__global__ __launch_bounds__(256) void cwl_partial_kernel(
    const float* __restrict__ lsig, const float* __restrict__ lrisk,
    const int* __restrict__ lab_s, const int* __restrict__ lab_r,
    float* __restrict__ partials, int B)
{
    const int tid    = threadIdx.x;
    const int gtid   = blockIdx.x * blockDim.x + tid;
    const int stride = gridDim.x * blockDim.x;
    const int nq     = B >> 2;          // row-quads: 12 floats = 3x b128 per task

    const f4* ls4 = (const f4*)lsig;
    const f4* lr4 = (const f4*)lrisk;
    const i4* cs4 = (const i4*)lab_s;
    const i4* cr4 = (const i4*)lab_r;

    float accS = 0.0f, accR = 0.0f;

    for (int q = gtid; q < nq; q += stride) {
        // single-pass data: non-temporal wide loads
        f4 a = __builtin_nontemporal_load(&ls4[3 * q + 0]);
        f4 b = __builtin_nontemporal_load(&ls4[3 * q + 1]);
        f4 c = __builtin_nontemporal_load(&ls4[3 * q + 2]);
        i4 ks = __builtin_nontemporal_load(&cs4[q]);
        accS += row_loss(a.x, a.y, a.z, ks.x);
        accS += row_loss(a.w, b.x, b.y, ks.y);
        accS += row_loss(b.z, b.w, c.x, ks.z);
        accS += row_loss(c.y, c.z, c.w, ks.w);

        f4 d = __builtin_nontemporal_load(&lr4[3 * q + 0]);
        f4 e = __builtin_nontemporal_load(&lr4[3 * q + 1]);
        f4 f = __builtin_nontemporal_load(&lr4[3 * q + 2]);
        i4 kr = __builtin_nontemporal_load(&cr4[q]);
        accR += row_loss(d.x, d.y, d.z, kr.x);
        accR += row_loss(d.w, e.x, e.y, kr.y);
        accR += row_loss(e.z, e.w, f.x, kr.z);
        accR += row_loss(f.y, f.z, f.w, kr.w);
    }

    // tail rows (B % 4), handled by block 0 only; reconverges before WMMA
    const int rem = B & 3;
    if (blockIdx.x == 0 && tid < rem) {
        int r = nq * 4 + tid;
        accS += row_loss(lsig[3 * r], lsig[3 * r + 1], lsig[3 * r + 2], lab_s[r]);
        accR += row_loss(lrisk[3 * r], lrisk[3 * r + 1], lrisk[3 * r + 2], lab_r[r]);
    }

    // ---- wave32 reduction via V_WMMA_F32_16X16X4_F32 ----
    // B-matrix (4x16 f32, 2 VGPRs): K0/K2 = accS lanes 0-15/16-31,
    //                               K1/K3 = accR lanes 0-15/16-31.
    // A-matrix selector: row0 = [1,1,0,0] (accS+accR, lanes 0-15), row1 = [0,0,1,1] (accS+accR, lanes 16-31).
    // Per 16x4 A layout: VGPR0 lane L holds (M=L,K=0 | M=L-16,K=2),
    //                    VGPR1 lane L holds (M=L,K=1 | M=L-16,K=3),
    // so lane 0 -> A[0][0]/A[0][1] = 1, lane 17 -> A[1][2]/A[1][3] = 1.
    const int lane = tid & 31;
    float sel = (lane == 0 || lane == 17) ? 1.0f : 0.0f;
    v2f Am; Am.x = sel;  Am.y = sel;
    v2f Bm; Bm.x = accS; Bm.y = accR;
    v8f Cm = {};
    v8f Dm = __builtin_amdgcn_wmma_f32_16x16x4_f32(
        /*neg_a=*/false, Am, /*neg_b=*/false, Bm,
        /*c_mod=*/(short)0, Cm, /*reuse_a=*/false, /*reuse_b=*/false);

    // D VGPR0 lanes 0-15: M=0 row = accS(n)+accR(n) per column n
    // D VGPR1 lanes 0-15: M=1 row = accS(n+16)+accR(n+16) per column n
    float sh = Dm[0];
    float rh = Dm[1];
    XOR_SWIZZLE_ADD(sh, 1); XOR_SWIZZLE_ADD(rh, 1);
    XOR_SWIZZLE_ADD(sh, 2); XOR_SWIZZLE_ADD(rh, 2);
    XOR_SWIZZLE_ADD(sh, 4); XOR_SWIZZLE_ADD(rh, 4);
    XOR_SWIZZLE_ADD(sh, 8); XOR_SWIZZLE_ADD(rh, 8);
    // lane 0 now holds accS+accR totals of lanes 0-15 (sh) and 16-31 (rh)

    __shared__ float smem[16];  // 8 waves x 2
    const int wave = tid >> 5;
    if (lane == 0) { smem[2 * wave] = sh; smem[2 * wave + 1] = rh; }
    __syncthreads();
    if (tid == 0) {
        float tS = 0.0f, tR = 0.0f;
        const int nw = blockDim.x >> 5;
        for (int w = 0; w < nw; ++w) { tS += smem[2 * w]; tR += smem[2 * w + 1]; }
        partials[2 * blockIdx.x]     = tS;
        partials[2 * blockIdx.x + 1] = tR;
    }
}

__global__ __launch_bounds__(256) void cwl_final_kernel(
    const float* __restrict__ partials, int nb, float* __restrict__ out, float invB)
{
    __shared__ float sS[256];
    __shared__ float sR[256];
    const int t = threadIdx.x;
    float aS = 0.0f, aR = 0.0f;
    for (int j = t; j < nb; j += 256) {  // fixed order -> deterministic
        aS += partials[2 * j];
        aR += partials[2 * j + 1];
    }
    sS[t] = aS; sR[t] = aR;
    __syncthreads();
    for (int off = 128; off > 0; off >>= 1) {
        if (t < off) { sS[t] += sS[t + off]; sR[t] += sR[t + off]; }
        __syncthreads();
    }
    if (t == 0) {
        float lossS = sS[0] * invB;
        float lossR = sR[0] * invB;
        out[0] = lossS;
        out[1] = lossR;
        out[2] = 1.0f * lossS + 0.5f * lossR;  // TASK_WEIGHTS
    }
}

extern "C" void kernel_launch(void* const* d_in, const int* in_sizes, int n_in,
                              void* d_out, int out_size, void* d_ws, size_t ws_size,
                              hipStream_t stream) {
    const float* lsig  = (const float*)d_in[0];
    const float* lrisk = (const float*)d_in[1];
    const int*   lab_s = (const int*)d_in[2];
    const int*   lab_r = (const int*)d_in[3];
    float* out      = (float*)d_out;
    float* partials = (float*)d_ws;

    const int B = in_sizes[2];  // labels_signal element count

    // one (sig,risk) pair per block in d_ws
    int nb = 2048;
    size_t max_nb = ws_size / (2 * sizeof(float));
    if ((size_t)nb > max_nb) nb = (int)max_nb;
    if (nb < 1) nb = 1;

    cwl_partial_kernel<<<nb, 256, 0, stream>>>(lsig, lrisk, lab_s, lab_r, partials, B);
    cwl_final_kernel<<<1, 256, 0, stream>>>(partials, nb, out, 1.0f / (float)B);
}